// IntegratedDeltaPoseLoss_2379411882401
// MI455X (gfx1250) — compile-verified
//
#include <hip/hip_runtime.h>

// IntegratedDeltaPoseLoss for MI455X (gfx1250).
// Bandwidth-bound (≈243 MB @ 23.3 TB/s ≈ 10.4 us floor); no WMMA applies
// (quat prefix-product is nonlinear; triangular-matmul cumsum would waste
// >100x FLOPs/bytes). CDNA5 paths used instead:
//   * GLOBAL_LOAD_ASYNC_TO_LDS_B128/B64 (+ s_wait_asynccnt) to stage each
//     28 KB row and overlap the target-row fetch with the predicted-row scan,
//     with th:TH_LOAD_NT so the once-read 235 MB stream does not thrash the
//     192 MB L2,
//   * wave32 ds_bpermute shuffles for a 2-barrier-per-input 3-level scan,
//   * non-temporal output stores (write-once 8 MB).

#define T_LEN  512
#define CH      14
#define NWAVES  16   // 512 threads / wave32

__device__ __forceinline__ void async_copy_b128(unsigned lds_addr, const float* gaddr) {
  asm volatile("global_load_async_to_lds_b128 %0, %1, off th:TH_LOAD_NT"
               : : "v"(lds_addr), "v"(gaddr)
               : "memory");
}

__device__ __forceinline__ void async_copy_b64(unsigned lds_addr, const float* gaddr) {
  asm volatile("global_load_async_to_lds_b64 %0, %1, off th:TH_LOAD_NT"
               : : "v"(lds_addr), "v"(gaddr)
               : "memory");
}

struct Pose {
  float lp[3];  // left position (cumsum)
  float lq[4];  // left quat xyzw (prefix product)
  float rp[3];  // right position
  float rq[4];  // right quat
};

// Hamilton product, xyzw; a = earlier prefix, b = later element.  r = a*b
__device__ __forceinline__ void quat_mul4(const float a[4], const float b[4], float r[4]) {
  r[0] = a[3]*b[0] + a[0]*b[3] + a[1]*b[2] - a[2]*b[1];
  r[1] = a[3]*b[1] - a[0]*b[2] + a[1]*b[3] + a[2]*b[0];
  r[2] = a[3]*b[2] + a[0]*b[1] - a[1]*b[0] + a[2]*b[3];
  r[3] = a[3]*b[3] - a[0]*b[0] - a[1]*b[1] - a[2]*b[2];
}

__device__ __forceinline__ Pose combine(const Pose& a, const Pose& b) {
  Pose r;
#pragma unroll
  for (int i = 0; i < 3; ++i) { r.lp[i] = a.lp[i] + b.lp[i]; r.rp[i] = a.rp[i] + b.rp[i]; }
  quat_mul4(a.lq, b.lq, r.lq);
  quat_mul4(a.rq, b.rq, r.rq);
  return r;
}

__device__ __forceinline__ Pose shfl_up_pose(const Pose& p, unsigned d) {
  Pose r;
#pragma unroll
  for (int i = 0; i < 3; ++i) { r.lp[i] = __shfl_up(p.lp[i], d); r.rp[i] = __shfl_up(p.rp[i], d); }
#pragma unroll
  for (int i = 0; i < 4; ++i) { r.lq[i] = __shfl_up(p.lq[i], d); r.rq[i] = __shfl_up(p.rq[i], d); }
  return r;
}

__device__ __forceinline__ void store_pose(float* dst, const Pose& p) {
#pragma unroll
  for (int i = 0; i < 3; ++i) dst[i]      = p.lp[i];
#pragma unroll
  for (int i = 0; i < 4; ++i) dst[3 + i]  = p.lq[i];
#pragma unroll
  for (int i = 0; i < 3; ++i) dst[7 + i]  = p.rp[i];
#pragma unroll
  for (int i = 0; i < 4; ++i) dst[10 + i] = p.rq[i];
}

__device__ __forceinline__ Pose load_pose(const float* src) {
  Pose p;
#pragma unroll
  for (int i = 0; i < 3; ++i) p.lp[i] = src[i];
#pragma unroll
  for (int i = 0; i < 4; ++i) p.lq[i] = src[3 + i];
#pragma unroll
  for (int i = 0; i < 3; ++i) p.rp[i] = src[7 + i];
#pragma unroll
  for (int i = 0; i < 4; ++i) p.rq[i] = src[10 + i];
  return p;
}

__device__ __forceinline__ void aa2quat(float x, float y, float z, float q[4]) {
  float ang = sqrtf(x * x + y * y + z * z);
  float h   = 0.5f * ang;
  float k   = sinf(h) / (ang + 1e-08f);  // axis/(|aa|+AA_EPS) * sin(half)
  q[0] = x * k; q[1] = y * k; q[2] = z * k; q[3] = cosf(h);
}

// quat_geodesic_angle with reference semantics: normalize by (||q||+1e-5),
// abs(dot), clip to <= 1-1e-5, (2*acos)^2.
__device__ __forceinline__ float quat_loss(const float a[4], const float b[4]) {
  float na = sqrtf(a[0]*a[0] + a[1]*a[1] + a[2]*a[2] + a[3]*a[3]) + 1e-05f;
  float nb = sqrtf(b[0]*b[0] + b[1]*b[1] + b[2]*b[2] + b[3]*b[3]) + 1e-05f;
  float d  = fabsf(a[0]*b[0] + a[1]*b[1] + a[2]*b[2] + a[3]*b[3]) / (na * nb);
  d = fminf(d, 1.0f - 1e-05f);
  float ang = 2.0f * acosf(d);
  return ang * ang;
}

// Three-level inclusive scan over T=512 (one thread per timestep):
//   5-step wave32 shuffle scan -> wave totals in LDS -> wave0 scans 16 totals
//   -> waves fold in their exclusive prefix.  Two barriers total.
__device__ __forceinline__ Pose scan_row(const float* stage, float* totals, float* pfx,
                                         int t, int lane, int wave,
                                         float& g0, float& g1) {
  const float* c = stage + t * CH;
  float v[CH];
#pragma unroll
  for (int i = 0; i < CH; ++i) v[i] = c[i];
  g0 = v[12]; g1 = v[13];  // grips stay raw deltas

  Pose p;
  p.lp[0] = v[0]; p.lp[1] = v[1]; p.lp[2] = v[2];
  p.rp[0] = v[6]; p.rp[1] = v[7]; p.rp[2] = v[8];
  aa2quat(v[3], v[4],  v[5],  p.lq);
  aa2quat(v[9], v[10], v[11], p.rq);

#pragma unroll
  for (unsigned o = 1; o < 32; o <<= 1) {
    Pose q = shfl_up_pose(p, o);
    if (lane >= (int)o) p = combine(q, p);
  }

  if (lane == 31) store_pose(totals + wave * CH, p);
  __syncthreads();

  if (wave == 0) {
    Pose w;
    if (lane < NWAVES) {
      w = load_pose(totals + lane * CH);
    } else {  // identity: zero translation, unit quaternion (unused lanes)
#pragma unroll
      for (int i = 0; i < 3; ++i) { w.lp[i] = 0.0f; w.rp[i] = 0.0f; }
      w.lq[0] = w.lq[1] = w.lq[2] = 0.0f; w.lq[3] = 1.0f;
      w.rq[0] = w.rq[1] = w.rq[2] = 0.0f; w.rq[3] = 1.0f;
    }
#pragma unroll
    for (unsigned o = 1; o < NWAVES; o <<= 1) {
      Pose q = shfl_up_pose(w, o);
      if (lane >= (int)o) w = combine(q, w);
    }
    if (lane < NWAVES) store_pose(pfx + lane * CH, w);
  }
  __syncthreads();

  if (wave > 0) {
    Pose pre = load_pose(pfx + (wave - 1) * CH);  // LDS broadcast read
    p = combine(pre, p);
  }
  return p;
}

__global__ __launch_bounds__(T_LEN)
void integrated_pose_loss_kernel(const float* __restrict__ pred,
                                 const float* __restrict__ tgt,
                                 float* __restrict__ out) {
  __shared__ float stageP[T_LEN * CH];   // 28 KB predicted row
  __shared__ float stageT[T_LEN * CH];   // 28 KB target row
  __shared__ float totals[NWAVES * CH];
  __shared__ float pfx[NWAVES * CH];

  const int t    = threadIdx.x;
  const int b    = blockIdx.x;
  const int lane = t & 31;
  const int wave = t >> 5;

  const size_t rowBase = (size_t)b * (T_LEN * CH);
  const float* gp = pred + rowBase;
  const float* gt = tgt  + rowBase;
  // Generic pointers into LDS carry the wave-relative LDS byte address in
  // their low 32 bits (aperture in the high bits) — exactly what the async
  // instruction's VDST expects.
  const unsigned lpA = (unsigned)(size_t)&stageP[0];
  const unsigned ltA = (unsigned)(size_t)&stageT[0];

  // Stage both rows asynchronously.  28 KB row = 512 threads x (3 float4 +
  // 1 float2): 3 x B128 + 1 x B64 async issues per thread, each instruction a
  // wave-contiguous chunk.  Row base b*28672 B and the float4 region are
  // 16 B aligned on both the global and LDS side.
#pragma unroll
  for (int k = 0; k < 3; ++k) {
    int i4 = t + T_LEN * k;                    // float4 index within row
    async_copy_b128(lpA + (unsigned)i4 * 16u, gp + (size_t)i4 * 4);
  }
  async_copy_b64(lpA + 24576u + (unsigned)t * 8u, gp + 6144 + (size_t)t * 2);
#pragma unroll
  for (int k = 0; k < 3; ++k) {
    int i4 = t + T_LEN * k;
    async_copy_b128(ltA + (unsigned)i4 * 16u, gt + (size_t)i4 * 4);
  }
  async_copy_b64(ltA + 24576u + (unsigned)t * 8u, gt + 6144 + (size_t)t * 2);

  // Async loads complete in order: <=4 outstanding means the 4 predicted-row
  // ops of this wave are done; barrier makes all waves' data visible while
  // the target row is still streaming in behind the first scan.
  __builtin_amdgcn_s_wait_asynccnt(4);
  __syncthreads();

  float pg0, pg1, tg0, tg1;
  Pose P = scan_row(stageP, totals, pfx, t, lane, wave, pg0, pg1);

  __builtin_amdgcn_s_wait_asynccnt(0);
  __syncthreads();

  Pose Q = scan_row(stageT, totals, pfx, t, lane, wave, tg0, tg1);

  // pose_loss
  float pos = 0.0f;
#pragma unroll
  for (int i = 0; i < 3; ++i) {
    float dl = P.lp[i] - Q.lp[i];
    float dr = P.rp[i] - Q.rp[i];
    pos += dl * dl + dr * dr;
  }
  pos *= (1.0f / 3.0f);  // mean over 3 for each arm, summed

  float ql   = quat_loss(P.lq, Q.lq);
  float qr   = quat_loss(P.rq, Q.rq);
  float grip = fabsf(pg0 - tg0) + fabsf(pg1 - tg1);

  // Write-once output: keep it out of the cache hierarchy's RT set.
  __builtin_nontemporal_store(pos + ql + qr + grip, &out[(size_t)b * T_LEN + t]);
}

extern "C" void kernel_launch(void* const* d_in, const int* in_sizes, int n_in,
                              void* d_out, int out_size, void* d_ws, size_t ws_size,
                              hipStream_t stream) {
  const float* pred = (const float*)d_in[0];
  const float* tgt  = (const float*)d_in[1];
  float* out        = (float*)d_out;
  const int B = in_sizes[0] / (T_LEN * CH);  // 4096 for the reference shapes
  hipLaunchKernelGGL(integrated_pose_loss_kernel, dim3(B), dim3(T_LEN), 0, stream,
                     pred, tgt, out);
  (void)n_in; (void)out_size; (void)d_ws; (void)ws_size;
}